// chamfer_loss_35424890257893
// MI455X (gfx1250) — compile-verified
//
#include <hip/hip_runtime.h>

typedef __attribute__((ext_vector_type(2))) float v2f;
typedef __attribute__((ext_vector_type(8))) float v8f;

#define BATCH 4
#define NPTS  8192
#define TILES (NPTS / 16)               // 512 target tiles per batch
#define WAVES_PER_PASS (BATCH * TILES)  // 2048 waves per directed pass
#define TOTAL_WAVES (2 * WAVES_PER_PASS)
#define BLOCK_THREADS 256               // 8 wave32s per block

// ---------------------------------------------------------------------------
// Pack (p0,p1,p2) -> (-2p0, -2p1, -2p2, |p|^2).  This single form serves as:
//   B fragment (target role): pure aligned 8-byte load, zero VALU per tile
//   A fragment (query role):  recovered once per wave with two *(-0.5) ops
// ---------------------------------------------------------------------------
__global__ void chamfer_pack_kernel(const float* __restrict__ pred,
                                    const float* __restrict__ gt,
                                    float4* __restrict__ packPred,
                                    float4* __restrict__ packGt) {
  int i = blockIdx.x * blockDim.x + threadIdx.x;
  if (i < BATCH * NPTS) {
    float x0 = pred[i * 3 + 0], x1 = pred[i * 3 + 1], x2 = pred[i * 3 + 2];
    packPred[i] =
        make_float4(-2.f * x0, -2.f * x1, -2.f * x2, x0 * x0 + x1 * x1 + x2 * x2);
    float y0 = gt[i * 3 + 0], y1 = gt[i * 3 + 1], y2 = gt[i * 3 + 2];
    packGt[i] =
        make_float4(-2.f * y0, -2.f * y1, -2.f * y2, y0 * y0 + y1 * y1 + y2 * y2);
  }
}

// ---------------------------------------------------------------------------
// Both directed Chamfer passes in one launch (pass = gwave >> 11).
// Per 16x16 tile:  D = A * B + 0  via v_wmma_f32_16x16x4_f32 with
//   A row i = ( q0, q1, q2, 1 )            (loop-invariant)
//   B col j = (-2t0,-2t1,-2t2,|t|^2)       (one b64 load per lane per tile)
// D[i][j] = |t_j|^2 - 2 q_i . t_j ; the row-constant |q_i|^2 is added after
// the j-min (min_j(c + f(j)) == c + min_j f(j)), then clamped to 0.
// Hot loop per tile: 1 vmem load + 1 wmma + 8 v_min. No DS, no cndmask.
// ---------------------------------------------------------------------------
__global__ __launch_bounds__(BLOCK_THREADS) void
chamfer_main_kernel(const float* __restrict__ packPred,
                    const float* __restrict__ packGt,
                    float* __restrict__ partials)  // [TOTAL_WAVES]
{
  const int lane  = threadIdx.x & 31;
  const int gwave = blockIdx.x * (BLOCK_THREADS >> 5) + (threadIdx.x >> 5);
  const int pass  = gwave >> 11;                 // 0: pred->gt, 1: gt->pred
  const int wave  = gwave & (WAVES_PER_PASS - 1);
  const float* Q  = pass ? packGt : packPred;
  const float* T  = pass ? packPred : packGt;

  const int b     = wave / TILES;                // uniform per wave
  const int itile = wave % TILES;
  const int l16   = lane & 15;
  const bool hi   = lane >= 16;                  // lanes 16-31 hold K = 2,3

  // --- A fragment (once per wave) -------------------------------------------
  // lo lanes load (-2q0,-2q1) -> (q0,q1); hi lanes load (-2q2,|q|^2) -> (q2,1)
  v2f ae = *(const v2f*)(Q + (size_t)(b * NPTS + itile * 16 + l16) * 4 +
                         (hi ? 2 : 0));
  const float rownorm = ae.y;                    // hi lane 16+m: |q_{base+m}|^2
  v2f a;
  a.x = -0.5f * ae.x;
  a.y = hi ? 1.0f : -0.5f * ae.y;

  v8f mn;
#pragma unroll
  for (int r = 0; r < 8; ++r) mn[r] = 3.0e38f;

  v8f cz = {};  // C = 0

  const float* tbase = T + (size_t)b * NPTS * 4 + (hi ? 2 : 0);

#pragma unroll 8
  for (int jt = 0; jt < TILES; ++jt) {
    v2f bf = *(const v2f*)(tbase + (size_t)(jt * 16 + l16) * 4);
    // 8 args: (neg_a, A, neg_b, B, c_mod, C, reuse_a, reuse_b)
    v8f d = __builtin_amdgcn_wmma_f32_16x16x4_f32(
        false, a, false, bf, (short)0, cz, false, false);
#pragma unroll
    for (int r = 0; r < 8; ++r) mn[r] = fminf(mn[r], d[r]);
  }

  // Min across the 16-lane half: covers all M columns per row.
#pragma unroll
  for (int off = 1; off < 16; off <<= 1) {
#pragma unroll
    for (int r = 0; r < 8; ++r)
      mn[r] = fminf(mn[r], __shfl_xor(mn[r], off, 32));
  }

  // Add per-row |q|^2 (held in hi-half lanes' A load), clamp, and sum.
  // Lo half holds rows 0-7 (norms at lanes 16..23), hi half rows 8-15
  // (norms at lanes 24..31).
  const int rb = hi ? 8 : 0;
  float s = 0.0f;
#pragma unroll
  for (int r = 0; r < 8; ++r) {
    float nrm = __shfl(rownorm, 16 + rb + r, 32);
    s += fmaxf(mn[r] + nrm, 0.0f);
  }
  s += __shfl_xor(s, 16, 32);                    // combine the two halves

  if (lane == 0) partials[gwave] = s;
}

// ---------------------------------------------------------------------------
// Deterministic final reduction: fixed-order strided sums + LDS tree.
// loss = (sum_x + sum_y) / (BATCH * NPTS)   (N == M, both means identical)
// ---------------------------------------------------------------------------
__global__ __launch_bounds__(256) void
chamfer_reduce_kernel(const float* __restrict__ partials,
                      float* __restrict__ out) {
  __shared__ float sm[256];
  int t = threadIdx.x;
  float s = 0.0f;
  for (int i = t; i < TOTAL_WAVES; i += 256) s += partials[i];
  sm[t] = s;
  __syncthreads();
  for (int off = 128; off > 0; off >>= 1) {
    if (t < off) sm[t] += sm[t + off];
    __syncthreads();
  }
  if (t == 0) out[0] = sm[0] / (float)(BATCH * NPTS);
}

extern "C" void kernel_launch(void* const* d_in, const int* in_sizes, int n_in,
                              void* d_out, int out_size, void* d_ws,
                              size_t ws_size, hipStream_t stream) {
  const float* pred = (const float*)d_in[0];
  const float* gt   = (const float*)d_in[1];

  float* ws       = (float*)d_ws;
  float* packPred = ws;                                 // BATCH*NPTS*4 floats
  float* packGt   = ws + (size_t)BATCH * NPTS * 4;      // BATCH*NPTS*4 floats
  float* partials = ws + (size_t)2 * BATCH * NPTS * 4;  // TOTAL_WAVES floats

  int npack = BATCH * NPTS;
  chamfer_pack_kernel<<<(npack + 255) / 256, 256, 0, stream>>>(
      pred, gt, (float4*)packPred, (float4*)packGt);

  dim3 grid(TOTAL_WAVES / (BLOCK_THREADS / 32));  // 512 blocks, both passes
  chamfer_main_kernel<<<grid, BLOCK_THREADS, 0, stream>>>(packPred, packGt,
                                                          partials);

  chamfer_reduce_kernel<<<1, 256, 0, stream>>>(partials, (float*)d_out);
}